// GIN2_31602369364484
// MI455X (gfx1250) — compile-verified
//
#include <hip/hip_runtime.h>

// GIN inference, MI455X (gfx1250, wave32).
// Strategy: push each layer's W1 GEMM *before* the edge aggregation
// (segment_sum is linear), so all edge traffic is 16 floats/edge.
// All matmuls run on V_WMMA_F32_16X16X4_F32 (exact fp32).
// Intermediates (N x 16 f32 = 6.4 MB) are L2-resident on the 192 MB L2.

#define N_NODES 100000
#define N_EDGES 3200000
#define F_INF   128
#define NT      (N_NODES / 16)   // 6250 node tiles (N divisible by 16)
#define BN_EPS  1e-5f

typedef float v2f __attribute__((ext_vector_type(2)));
typedef float v8f __attribute__((ext_vector_type(8)));

// D = A(16x4 f32) * B(4x16 f32) + C(16x16 f32), wave32.
// Fragment layout (ISA 7.12.2, 32-bit operands):
//   A: lane(half,m): v0 = A[M=m][K=2*half+0], v1 = A[M=m][K=2*half+1]
//   B: lane(half,m): v0 = B[K=2*half+0][N=m], v1 = B[K=2*half+1][N=m]
//   C/D: lane(half,m), reg r: C[M = r + 8*half][N = m]
__device__ __forceinline__ v8f wmma_f32(v2f a, v2f b, v8f c) {
  return __builtin_amdgcn_wmma_f32_16x16x4_f32(false, a, false, b,
                                               (short)0, c, false, false);
}

// ---------------------------------------------------------------- conv1 GEMM
// y[N,16] = x[N,128] @ W[128,16]   (bias folded in later, with agg)
__global__ void k_gemm_x_w1(const float* __restrict__ x,
                            const float* __restrict__ W,
                            float* __restrict__ y) {
  const int lane = threadIdx.x & 31;
  const int half = lane >> 4, m = lane & 15;
  const int tile = blockIdx.x * (blockDim.x >> 5) + (threadIdx.x >> 5);
  if (tile >= NT) return;            // wave-uniform guard
  const int nb = tile * 16;
  const float* xrow = x + (size_t)(nb + m) * F_INF;
  v8f acc = {};
  for (int k0 = 0; k0 < F_INF; k0 += 4) {
    v2f a = *(const v2f*)(xrow + k0 + 2 * half);
    v2f b;
    b.x = W[(size_t)(k0 + 2 * half + 0) * 16 + m];
    b.y = W[(size_t)(k0 + 2 * half + 1) * 16 + m];
    acc = wmma_f32(a, b, acc);
  }
  float* yt = y + (size_t)nb * 16;
#pragma unroll
  for (int r = 0; r < 8; ++r) yt[(size_t)(r + 8 * half) * 16 + m] = acc[r];
}

// ---------------------------------------------------------------- zero
__global__ void k_zero4(float4* __restrict__ p, int n4) {
  int i = blockIdx.x * blockDim.x + threadIdx.x;
  if (i < n4) p[i] = make_float4(0.f, 0.f, 0.f, 0.f);
}

// ---------------------------------------------------------------- scatter-add
// t[col[e]] += y[row[e]]  (16 f32/edge; 4 features per thread; L2-resident)
__global__ void k_scatter(const float* __restrict__ y,
                          const int* __restrict__ row,
                          const int* __restrict__ col,
                          float* __restrict__ t) {
  int idx = blockIdx.x * blockDim.x + threadIdx.x;
  int e = idx >> 2, q = (idx & 3) * 4;
  if (e >= N_EDGES) return;
  int r = row[e], c = col[e];
  float4 v = *(const float4*)(y + (size_t)r * 16 + q);
  float* dst = t + (size_t)c * 16 + q;
  atomicAdd(dst + 0, v.x);
  atomicAdd(dst + 1, v.y);
  atomicAdd(dst + 2, v.z);
  atomicAdd(dst + 3, v.w);
}

// ---------------------------------------------------------------- node MLP
// z = relu(BN(y + t + b1));  h = relu(z @ W2 + b2);
// yout = h @ Wn            (relu_out == 0, Wn = next layer's W1)
// yout = relu(h @ Wn + bn2)(relu_out == 1, Wn/bn2 = lin1 W/b)
__global__ void k_node_mlp(const float* __restrict__ y,
                           const float* __restrict__ t,
                           const float* __restrict__ b1,
                           const float* __restrict__ gamma,
                           const float* __restrict__ beta,
                           const float* __restrict__ mean,
                           const float* __restrict__ var,
                           const float* __restrict__ W2,
                           const float* __restrict__ b2,
                           const float* __restrict__ Wn,
                           const float* __restrict__ bn2,
                           int relu_out,
                           float* __restrict__ yout) {
  __shared__ float hs[8][16][17];      // per-wave 16x16 tile, padded rows
  const int wband = threadIdx.x >> 5;
  const int lane = threadIdx.x & 31;
  const int half = lane >> 4, m = lane & 15;
  const int tile = blockIdx.x * (blockDim.x >> 5) + wband;
  const bool active = (tile < NT);

  if (active) {
    const float* yrow = y + (size_t)(tile * 16 + m) * 16;
    const float* trow = t + (size_t)(tile * 16 + m) * 16;
    v8f acc = {};
#pragma unroll
    for (int ks = 0; ks < 4; ++ks) {
      const int f = ks * 4 + 2 * half;
      v2f yv = *(const v2f*)(yrow + f);
      v2f tv = *(const v2f*)(trow + f);
      v2f bv = *(const v2f*)(b1 + f);
      v2f gv = *(const v2f*)(gamma + f);
      v2f be = *(const v2f*)(beta + f);
      v2f mv = *(const v2f*)(mean + f);
      v2f vv = *(const v2f*)(var + f);
      v2f a;
      float s0 = gv.x * __frsqrt_rn(vv.x + BN_EPS);
      float s1 = gv.y * __frsqrt_rn(vv.y + BN_EPS);
      a.x = fmaxf(0.f, (yv.x + tv.x + bv.x - mv.x) * s0 + be.x);
      a.y = fmaxf(0.f, (yv.y + tv.y + bv.y - mv.y) * s1 + be.y);
      v2f b;
      b.x = W2[(size_t)(f + 0) * 16 + m];
      b.y = W2[(size_t)(f + 1) * 16 + m];
      acc = wmma_f32(a, b, acc);
    }
    const float bias2 = b2[m];
#pragma unroll
    for (int r = 0; r < 8; ++r)
      hs[wband][r + 8 * half][m] = fmaxf(0.f, acc[r] + bias2);
  }
  __syncthreads();                     // D-layout -> A-layout via LDS
  if (active) {
    v8f acc2 = {};
#pragma unroll
    for (int ks = 0; ks < 4; ++ks) {
      const int f = ks * 4 + 2 * half;
      v2f a, b;
      a.x = hs[wband][m][f + 0];
      a.y = hs[wband][m][f + 1];
      b.x = Wn[(size_t)(f + 0) * 16 + m];
      b.y = Wn[(size_t)(f + 1) * 16 + m];
      acc2 = wmma_f32(a, b, acc2);
    }
    float* yt = yout + (size_t)tile * 16 * 16;
#pragma unroll
    for (int r = 0; r < 8; ++r) {
      float v = acc2[r];
      if (relu_out) v = fmaxf(0.f, v + bn2[m]);
      yt[(size_t)(r + 8 * half) * 16 + m] = v;
    }
  }
}

// ---------------------------------------------------------------- edge output
// e_vec = [h[row], h[col]] (E x 32);  out = e_vec @ Wl2(32x6) + bl2 (E x 6)
// Pure streaming-store bound: E*38*4 B = 486 MB -> ~21 us @ 23.3 TB/s floor.
__global__ void k_edge_out(const float* __restrict__ h,
                           const int* __restrict__ row,
                           const int* __restrict__ col,
                           const float* __restrict__ W,   // [32,6]
                           const float* __restrict__ b,   // [6]
                           float* __restrict__ out,       // [E,6]
                           float* __restrict__ evec) {    // [E,32]
  __shared__ float Ws[32 * 6 + 6];
  for (int i = threadIdx.x; i < 198; i += blockDim.x)
    Ws[i] = (i < 192) ? W[i] : b[i - 192];
  __syncthreads();
  int e = blockIdx.x * blockDim.x + threadIdx.x;
  if (e >= N_EDGES) return;
  const int r = row[e], c = col[e];
  float ev[32];
  const float4* hr = (const float4*)(h + (size_t)r * 16);
  const float4* hc = (const float4*)(h + (size_t)c * 16);
#pragma unroll
  for (int q = 0; q < 4; ++q) {
    float4 a = hr[q], bb = hc[q];
    ev[q * 4 + 0] = a.x;  ev[q * 4 + 1] = a.y;
    ev[q * 4 + 2] = a.z;  ev[q * 4 + 3] = a.w;
    ev[16 + q * 4 + 0] = bb.x;  ev[16 + q * 4 + 1] = bb.y;
    ev[16 + q * 4 + 2] = bb.z;  ev[16 + q * 4 + 3] = bb.w;
  }
  float4* ed = (float4*)(evec + (size_t)e * 32);
#pragma unroll
  for (int q = 0; q < 8; ++q)
    ed[q] = make_float4(ev[q * 4], ev[q * 4 + 1], ev[q * 4 + 2], ev[q * 4 + 3]);
  float* op = out + (size_t)e * 6;
#pragma unroll
  for (int cc = 0; cc < 6; ++cc) {
    float s = Ws[192 + cc];
#pragma unroll
    for (int f = 0; f < 32; ++f) s = fmaf(ev[f], Ws[f * 6 + cc], s);
    op[cc] = s;
  }
}

// ---------------------------------------------------------------- launch
extern "C" void kernel_launch(void* const* d_in, const int* in_sizes, int n_in,
                              void* d_out, int out_size, void* d_ws, size_t ws_size,
                              hipStream_t stream) {
  // Inputs flattened in setup_inputs() insertion order:
  // 0: x [N,128]   1: edge_index [2,E] (int32)   2: batch [N] (unused)
  // 3..10:  conv1 {W1,b1,bn_gamma,bn_beta,bn_mean,bn_var,W2,b2}
  // 11..18: conv2, 19..26: conv3, 27..34: conv4
  // 35,36: lin1 {W,b}    37,38: lin2 {W,b}
  const float* x = (const float*)d_in[0];
  const int* row = (const int*)d_in[1];
  const int* col = row + N_EDGES;
  auto F = [&](int i) { return (const float*)d_in[i]; };

  float* y_a = (float*)d_ws;                       // N x 16
  float* y_b = y_a + (size_t)N_NODES * 16;         // N x 16
  float* t   = y_b + (size_t)N_NODES * 16;         // N x 16
  float* hf  = t   + (size_t)N_NODES * 16;         // N x 16
  float* out  = (float*)d_out;                     // E x 6
  float* evec = out + (size_t)N_EDGES * 6;         // E x 32

  const dim3 blk(256);
  const int gemm_blocks = (NT + 7) / 8;            // 8 waves/block, 1 tile/wave
  const int zero_blocks = (N_NODES * 16 / 4 + 255) / 256;
  const int scat_blocks = (N_EDGES * 4 + 255) / 256;
  const int edge_blocks = (N_EDGES + 255) / 256;

  // conv1 W1 GEMM before aggregation (linearity trick: 128-wide -> 16-wide edges)
  k_gemm_x_w1<<<gemm_blocks, blk, 0, stream>>>(x, F(3), y_a);

  float* ycur = y_a;
  float* ynext = y_b;
  for (int l = 0; l < 4; ++l) {
    const int p = 3 + l * 8;  // this conv's param base
    k_zero4<<<zero_blocks, blk, 0, stream>>>((float4*)t, N_NODES * 16 / 4);
    k_scatter<<<scat_blocks, blk, 0, stream>>>(ycur, row, col, t);
    if (l < 3) {
      // fused: elementwise+BN+ReLU -> W2 GEMM -> ReLU -> next layer's W1 GEMM
      k_node_mlp<<<gemm_blocks, blk, 0, stream>>>(
          ycur, t, F(p + 1), F(p + 2), F(p + 3), F(p + 4), F(p + 5),
          F(p + 6), F(p + 7), F(p + 9) /* next W1 */, F(36) /*unused*/, 0, ynext);
      float* tmp = ycur; ycur = ynext; ynext = tmp;
    } else {
      // last conv: fuse lin1 (W,b,ReLU) into the tail GEMM
      k_node_mlp<<<gemm_blocks, blk, 0, stream>>>(
          ycur, t, F(p + 1), F(p + 2), F(p + 3), F(p + 4), F(p + 5),
          F(p + 6), F(p + 7), F(35), F(36), 1, hf);
    }
  }

  k_edge_out<<<edge_blocks, blk, 0, stream>>>(hf, row, col, F(37), F(38),
                                              out, evec);
}